// EquilibriumResidualLoss_12146167513818
// MI455X (gfx1250) — compile-verified
//
#include <hip/hip_runtime.h>
#include <hip/hip_bf16.h>

// -----------------------------------------------------------------------------
// EquilibriumResidualLoss for MI455X (gfx1250), round 2.
//
// Roofline: ~0.2 GFLOP vs ~330 MB minimal traffic -> HBM/L2-atomic bound.
// Plan:
//   * u (=pred*J, 24 MB) and F_int (24 MB) staged in d_ws, resident in the
//     192 MB L2 -> 8M random gathers + 24M scatter atomics stay on-chip.
//   * stream-once data (element arrays, F_ext, bc masks, pred) loaded with
//     non-temporal hints (th:TH_LOAD_NT) so it can't evict the staged state.
//   * b128 vectorized streaming (4 nodes / 12 floats per thread in the
//     reduction; 4 floats per thread in node prep; b64 connectivity).
//   * scatter: no-return GLOBAL_ATOMIC_ADD_F32, relaxed / agent scope.
//   * reduction: wave32 shfl_xor butterfly -> LDS (8 waves) -> 1 atomic/block.
// -----------------------------------------------------------------------------

typedef float v4f __attribute__((ext_vector_type(4)));
typedef int   v2i __attribute__((ext_vector_type(2)));

#define NT_LD(p) __builtin_nontemporal_load(p)

static __device__ __forceinline__ void atomic_add_noret(float* addr, float v) {
    (void)__hip_atomic_fetch_add(addr, v, __ATOMIC_RELAXED, __HIP_MEMORY_SCOPE_AGENT);
}

// Kernel 0: u = pred*J (b128 path); F_int = 0; zero accumulators.
// Re-initializes all workspace every call -> graph-replay safe.
__global__ void __launch_bounds__(256)
eq_node_prep(const float* __restrict__ pred,
             const float* __restrict__ J,
             float* __restrict__ u,
             float* __restrict__ Fint,
             float* __restrict__ accum,
             int n3) {
    int i = blockIdx.x * blockDim.x + threadIdx.x;
    if (i < 2) accum[i] = 0.0f;

    const int nq = n3 >> 2;                      // quads of floats
    if (i < nq) {
        v4f p = NT_LD((const v4f*)pred + i);     // pred: read exactly once
        v4f j = *((const v4f*)J + i);            // J re-read in kernel 2: RT
        v4f z = {0.0f, 0.0f, 0.0f, 0.0f};
        *((v4f*)u + i)    = p * j;               // stays hot in L2
        *((v4f*)Fint + i) = z;
    }
    const int rem = n3 - (nq << 2);              // scalar tail (n3 % 4)
    if (i < rem) {
        int idx = (nq << 2) + i;
        u[idx]    = NT_LD(&pred[idx]) * J[idx];
        Fint[idx] = 0.0f;
    }
}

// Kernel 1: per-element local beam forces + scatter-add into F_int.
__global__ void __launch_bounds__(256)
eq_elem_scatter(const int*   __restrict__ conn,
                const float* __restrict__ len,
                const float* __restrict__ E,
                const float* __restrict__ A,
                const float* __restrict__ I22,
                const float* __restrict__ dir,
                const float* __restrict__ u,
                float* __restrict__ Fint,
                int nElem) {
    int e = blockIdx.x * blockDim.x + threadIdx.x;
    if (e >= nElem) return;

    // Streaming element data: non-temporal (single use, protect L2).
    v2i   nn = NT_LD((const v2i*)conn + e);      // one b64: (nA, nB)
    int   nA = nn.x, nB = nn.y;
    float c  = NT_LD(&dir[3 * e]);
    float s  = NT_LD(&dir[3 * e + 2]);
    float L  = NT_LD(&len[e]);
    float Ee = NT_LD(&E[e]);
    float Ae = NT_LD(&A[e]);
    float Ie = NT_LD(&I22[e]);

    // Random gathers: u is L2-resident (24 MB << 192 MB L2).
    const float* uA = u + 3 * nA;
    const float* uB = u + 3 * nB;
    float uAx = uA[0], uAy = uA[1], uAz = uA[2];
    float uBx = uB[0], uBy = uB[1], uBz = uB[2];

    float u_A =  c * uAx + s * uAy;
    float w_A = -s * uAx + c * uAy;
    float thA = -uAz;
    float u_B =  c * uBx + s * uBy;
    float w_B = -s * uBx + c * uBy;
    float thB = -uBz;

    float rl    = 1.0f / L;
    float ea_l  = Ee * Ae * rl;
    float ei_l  = Ee * Ie * rl;
    float ei_l2 = ei_l * rl;
    float ei_l3 = ei_l2 * rl;

    float f0  = ea_l * (u_A - u_B);                          // AXIAL_WEIGHT = 1
    float dw  = w_A - w_B;
    float f1  = 12.0f * ei_l3 * dw + 6.0f * ei_l2 * (thA + thB);
    float f2  =  6.0f * ei_l2 * dw + ei_l * (4.0f * thA + 2.0f * thB);
    float f5  =  6.0f * ei_l2 * dw + ei_l * (2.0f * thA + 4.0f * thB);

    float fAx = c * f0 - s * f1;
    float fAy = s * f0 + c * f1;
    // fB = (-fAx, -fAy, -f5)

    float* FA = Fint + 3 * nA;
    float* FB = Fint + 3 * nB;
    atomic_add_noret(&FA[0],  fAx);
    atomic_add_noret(&FA[1],  fAy);
    atomic_add_noret(&FA[2], -f2);
    atomic_add_noret(&FB[0], -fAx);
    atomic_add_noret(&FB[1], -fAy);
    atomic_add_noret(&FB[2], -f5);
}

// Kernel 2: masked residual squared-norm + free-DOF count.
// 4 nodes per thread: 3x b128 per array, fully aligned (48 B node-quads).
__global__ void __launch_bounds__(256)
eq_reduce(const float* __restrict__ Fint,
          const float* __restrict__ Fext,
          const float* __restrict__ J,
          const float* __restrict__ bcd,
          const float* __restrict__ bcr,
          float* __restrict__ accum,
          int nNodes) {
    __shared__ float s_sq[8];
    __shared__ float s_fr[8];

    const int t      = blockIdx.x * blockDim.x + threadIdx.x;
    const int stride = gridDim.x * blockDim.x;
    const int nquads = nNodes >> 2;

    const v4f* FiQ = (const v4f*)Fint;
    const v4f* FeQ = (const v4f*)Fext;
    const v4f* JQ  = (const v4f*)J;
    const v4f* BdQ = (const v4f*)bcd;
    const v4f* BrQ = (const v4f*)bcr;

    float sq = 0.0f, fcnt = 0.0f;

    for (int q = t; q < nquads; q += stride) {
        v4f bd = NT_LD(BdQ + q);                 // bc masks: single use
        v4f br = NT_LD(BrQ + q);
        v4f a0 = FiQ[3 * q],           a1 = FiQ[3 * q + 1],           a2 = FiQ[3 * q + 2];   // F_int: L2 hit
        v4f b0 = NT_LD(FeQ + 3 * q),   b1 = NT_LD(FeQ + 3 * q + 1),   b2 = NT_LD(FeQ + 3 * q + 2);
        v4f c0 = JQ[3 * q],            c1 = JQ[3 * q + 1],            c2 = JQ[3 * q + 2];

        float fi[12], fe[12], jj[12];
        #pragma unroll
        for (int k = 0; k < 4; ++k) {
            fi[k] = a0[k]; fi[4 + k] = a1[k]; fi[8 + k] = a2[k];
            fe[k] = b0[k]; fe[4 + k] = b1[k]; fe[8 + k] = b2[k];
            jj[k] = c0[k]; jj[4 + k] = c1[k]; jj[8 + k] = c2[k];
        }
        #pragma unroll
        for (int k = 0; k < 4; ++k) {
            float fd = 1.0f - bd[k];
            float fr = 1.0f - br[k];
            float jx = jj[3 * k], jy = jj[3 * k + 1], jz = jj[3 * k + 2];
            float Rx = (fi[3 * k]     - fe[3 * k])     * fd * (jx * jx);
            float Ry = (fi[3 * k + 1] - fe[3 * k + 1]) * fd * (jy * jy);
            float Rz = (fi[3 * k + 2] - fe[3 * k + 2]) * fr * (jz * jz);
            sq   += Rx * Rx + Ry * Ry + Rz * Rz;
            fcnt += 2.0f * fd + fr;
        }
    }

    // Scalar tail (nNodes % 4).
    for (int n = (nquads << 2) + t; n < nNodes; n += stride) {
        float fd = 1.0f - NT_LD(&bcd[n]);
        float fr = 1.0f - NT_LD(&bcr[n]);
        float jx = J[3 * n], jy = J[3 * n + 1], jz = J[3 * n + 2];
        float Rx = (Fint[3 * n]     - NT_LD(&Fext[3 * n]))     * fd * (jx * jx);
        float Ry = (Fint[3 * n + 1] - NT_LD(&Fext[3 * n + 1])) * fd * (jy * jy);
        float Rz = (Fint[3 * n + 2] - NT_LD(&Fext[3 * n + 2])) * fr * (jz * jz);
        sq   += Rx * Rx + Ry * Ry + Rz * Rz;
        fcnt += 2.0f * fd + fr;
    }

    // wave32 butterfly reduction.
    #pragma unroll
    for (int off = 16; off > 0; off >>= 1) {
        sq   += __shfl_xor(sq,   off, 32);
        fcnt += __shfl_xor(fcnt, off, 32);
    }
    int lane = threadIdx.x & 31;
    int wave = threadIdx.x >> 5;
    if (lane == 0) { s_sq[wave] = sq; s_fr[wave] = fcnt; }
    __syncthreads();
    if (wave == 0) {
        sq   = (lane < 8) ? s_sq[lane] : 0.0f;
        fcnt = (lane < 8) ? s_fr[lane] : 0.0f;
        #pragma unroll
        for (int off = 4; off > 0; off >>= 1) {
            sq   += __shfl_xor(sq,   off, 32);
            fcnt += __shfl_xor(fcnt, off, 32);
        }
        if (lane == 0) {
            atomic_add_noret(&accum[0], sq);
            atomic_add_noret(&accum[1], fcnt);
        }
    }
}

// Kernel 3: loss = sum_sq / max(n_free, 1)
__global__ void eq_finalize(const float* __restrict__ accum,
                            float* __restrict__ out) {
    if (blockIdx.x == 0 && threadIdx.x == 0) {
        out[0] = accum[0] / fmaxf(accum[1], 1.0f);
    }
}

extern "C" void kernel_launch(void* const* d_in, const int* in_sizes, int n_in,
                              void* d_out, int out_size, void* d_ws, size_t ws_size,
                              hipStream_t stream) {
    const float* pred = (const float*)d_in[0];   // (N_NODES,3)
    const float* J    = (const float*)d_in[1];   // (N_NODES,3)
    const int*   conn = (const int*)  d_in[2];   // (N_ELEM,2)
    const float* len  = (const float*)d_in[3];   // (N_ELEM,)
    const float* E    = (const float*)d_in[4];
    const float* A    = (const float*)d_in[5];
    const float* I22  = (const float*)d_in[6];
    const float* dir  = (const float*)d_in[7];   // (N_ELEM,3)
    const float* Fext = (const float*)d_in[8];   // (N_NODES,3)
    const float* bcd  = (const float*)d_in[9];   // (N_NODES,1)
    const float* bcr  = (const float*)d_in[10];  // (N_NODES,1)
    float* out = (float*)d_out;

    const int n3     = in_sizes[0];              // 3*N_NODES
    const int nNodes = in_sizes[9];              // N_NODES
    const int nElem  = in_sizes[3];              // N_ELEM

    // Workspace layout: u | F_int | accum[2]
    float* u     = (float*)d_ws;
    float* Fint  = u + n3;
    float* accum = Fint + n3;

    const int BLK = 256;                         // 8 wave32s per block

    const int prepThreads = (n3 + 3) / 4;        // one quad per thread (+tail)
    eq_node_prep<<<(prepThreads + BLK - 1) / BLK, BLK, 0, stream>>>(
        pred, J, u, Fint, accum, n3);

    eq_elem_scatter<<<(nElem + BLK - 1) / BLK, BLK, 0, stream>>>(
        conn, len, E, A, I22, dir, u, Fint, nElem);

    eq_reduce<<<2048, BLK, 0, stream>>>(
        Fint, Fext, J, bcd, bcr, accum, nNodes);

    eq_finalize<<<1, 32, 0, stream>>>(accum, out);
}